// Codebook_75849122447889
// MI455X (gfx1250) — compile-verified
//
#include <hip/hip_runtime.h>
#include <hip/hip_bf16.h>
#include <math.h>

typedef __attribute__((ext_vector_type(2))) float v2f;
typedef __attribute__((ext_vector_type(8))) float v8f;

#define NUM_EMB   1024
#define EMB_DIM   64
#define N_ROWS    65536     // B*H*W = 16*64*64
#define CHW       262144    // C*H*W = 64*4096
#define HW        4096      // H*W

// ---------------- K0: ||e||^2 per code, zero counts + loss accumulator ----
__global__ void k0_init(const float* __restrict__ emb,
                        float* __restrict__ eNormSq,
                        unsigned* __restrict__ counts,
                        float* __restrict__ lossAcc) {
  int t = blockIdx.x * blockDim.x + threadIdx.x;
  if (t < NUM_EMB) {
    const float* r = emb + (size_t)t * EMB_DIM;
    float s = 0.f;
    #pragma unroll
    for (int c = 0; c < EMB_DIM; ++c) { float v = r[c]; s += v * v; }
    eNormSq[t] = s;
    counts[t] = 0u;
  }
  if (t == 0) *lossAcc = 0.f;
}

// ---------------- K1: fp32 WMMA distances + argmin -----------------------
// One wave = one 16-row tile. 64 column-tiles of 16 codes; K=64 via 16
// accumulating v_wmma_f32_16x16x4_f32. Distance = ||e||^2 - 2*(z.e)
// (||z||^2 dropped: constant per row, argmin-invariant).
__global__ void __launch_bounds__(256)
k1_argmin(const float* __restrict__ x,
          const float* __restrict__ emb,
          const float* __restrict__ eNormSq,
          int* __restrict__ idxOut) {
  const int lane    = threadIdx.x & 31;
  const int wave    = threadIdx.x >> 5;
  const int tileRow = (blockIdx.x * 8 + wave) * 16;
  const int m       = lane & 15;       // row within tile (A), col within tile (B/C/D)
  const int half    = lane >> 4;       // K-half for A/B fragments

  // A fragment: row n = tileRow+m of flat [N,C]; flat[n][c] = x[b*CHW + c*HW + hw]
  const int n  = tileRow + m;
  const int b  = n >> 12;
  const int hw = n & 4095;
  const float* xb = x + (size_t)b * CHW + hw;

  v2f a[16];
  #pragma unroll
  for (int s = 0; s < 16; ++s) {
    const int c0 = 4 * s + half * 2;
    a[s].x = xb[(size_t)c0 * HW];
    a[s].y = xb[(size_t)(c0 + 1) * HW];
  }

  float bestD[8];
  int   bestI[8];
  #pragma unroll
  for (int r = 0; r < 8; ++r) { bestD[r] = __builtin_inff(); bestI[r] = 0x7fffffff; }

  for (int j = 0; j < NUM_EMB; j += 16) {
    v8f acc = {0.f, 0.f, 0.f, 0.f, 0.f, 0.f, 0.f, 0.f};
    // B fragment: B[c][code] = emb[code][c]; lane m -> code j+m, K split by half
    const float* brow = emb + (size_t)(j + m) * EMB_DIM + half * 2;
    #pragma unroll
    for (int s = 0; s < 16; ++s) {
      v2f bf = *reinterpret_cast<const v2f*>(brow + 4 * s);
      acc = __builtin_amdgcn_wmma_f32_16x16x4_f32(
              /*neg_a=*/false, a[s], /*neg_b=*/false, bf,
              /*c_mod=*/(short)0, acc, /*reuse_a=*/false, /*reuse_b=*/false);
    }
    const float en      = eNormSq[j + m];
    const int   codeIdx = j + m;
    #pragma unroll
    for (int r = 0; r < 8; ++r) {
      float d = en - 2.0f * acc[r];   // row = r + 8*half, col = m
      if (d < bestD[r]) { bestD[r] = d; bestI[r] = codeIdx; }   // strict < keeps lowest idx
    }
  }

  // Reduce each row's min across the 16 column-lanes (within each wave half),
  // lowest-index tie-break to match jnp.argmin.
  #pragma unroll
  for (int r = 0; r < 8; ++r) {
    float d = bestD[r];
    int   i = bestI[r];
    #pragma unroll
    for (int off = 1; off < 16; off <<= 1) {
      float od = __shfl_xor(d, off, 16);
      int   oi = __shfl_xor(i, off, 16);
      if (od < d || (od == d && oi < i)) { d = od; i = oi; }
    }
    if (m == 0) idxOut[tileRow + r + half * 8] = i;
  }
}

// ---------------- K2: gather + transposed write + loss + histogram -------
__global__ void __launch_bounds__(256)
k2_gather(const float* __restrict__ x,
          const float* __restrict__ emb,
          const int* __restrict__ idxArr,
          float* __restrict__ quantOut,      // [B,C,H,W]
          unsigned* __restrict__ counts,
          float* __restrict__ lossAcc) {
  __shared__ float red[256];
  const int t  = threadIdx.x;
  const int n  = blockIdx.x * 64 + (t & 63); // 64 consecutive rows per block
  const int cG = t >> 6;                     // 4 thread-groups x 16 channels
  const int idx = idxArr[n];
  const int b  = n >> 12;
  const int hw = n & 4095;
  const float* erow = emb + (size_t)idx * EMB_DIM + cG * 16;
  const float* xb   = x        + (size_t)b * CHW + hw;
  float*       qb   = quantOut + (size_t)b * CHW + hw;

  float s = 0.f;
  #pragma unroll
  for (int cc = 0; cc < 16; ++cc) {
    const int c = cG * 16 + cc;
    const float  e   = erow[cc];
    const size_t off = (size_t)c * HW;
    const float  dif = e - xb[off];
    qb[off] = e;                              // quantized_st == quantized numerically
    s += dif * dif;
  }
  if (cG == 0) atomicAdd(&counts[idx], 1u);

  red[t] = s;
  __syncthreads();
  for (int st = 128; st > 0; st >>= 1) {
    if (t < st) red[t] += red[t + st];
    __syncthreads();
  }
  if (t == 0) atomicAdd(lossAcc, red[0]);
}

// ---------------- K3: stream-zero the 256MB one-hot region ---------------
__global__ void k3_zero(float* __restrict__ enc, long n2) {
  const long stride = (long)gridDim.x * blockDim.x;
  const v2f z = {0.f, 0.f};
  for (long i = (long)blockIdx.x * blockDim.x + threadIdx.x; i < n2; i += stride) {
    __builtin_nontemporal_store(z, reinterpret_cast<v2f*>(enc) + i);
  }
}

// ---------------- K4: scatter ones --------------------------------------
__global__ void k4_scatter(const int* __restrict__ idxArr, float* __restrict__ enc) {
  const int n = blockIdx.x * 256 + threadIdx.x;
  enc[(size_t)n * NUM_EMB + idxArr[n]] = 1.0f;
}

// ---------------- K5: perplexity + loss finalize -------------------------
__global__ void k5_final(const unsigned* __restrict__ counts,
                         const float* __restrict__ lossAcc,
                         float* __restrict__ outLoss,
                         float* __restrict__ outPpl) {
  __shared__ float red[256];
  const int t = threadIdx.x;
  float s = 0.f;
  for (int i = t; i < NUM_EMB; i += 256) {
    const float p = (float)counts[i] * (1.0f / (float)N_ROWS);
    s += p * logf(p + 1e-10f);
  }
  red[t] = s;
  __syncthreads();
  for (int st = 128; st > 0; st >>= 1) {
    if (t < st) red[t] += red[t + st];
    __syncthreads();
  }
  if (t == 0) {
    *outPpl  = expf(-red[0]);
    *outLoss = 0.25f * (*lossAcc) / (float)(N_ROWS * EMB_DIM);
  }
}

extern "C" void kernel_launch(void* const* d_in, const int* in_sizes, int n_in,
                              void* d_out, int out_size, void* d_ws, size_t ws_size,
                              hipStream_t stream) {
  const float* x   = (const float*)d_in[0];   // [16,64,64,64]
  const float* emb = (const float*)d_in[1];   // [1024,64]

  // d_out: loss(1) | quantized_st [B,C,H,W] (4194304) | perplexity(1) | encodings (67108864)
  float* out     = (float*)d_out;
  float* outLoss = out;
  float* quant   = out + 1;
  float* outPpl  = out + 1 + (N_ROWS * EMB_DIM);
  float* enc     = out + 2 + (N_ROWS * EMB_DIM);

  char* ws = (char*)d_ws;
  int*      idxArr  = (int*)ws;                                 // 65536 ints
  float*    eNormSq = (float*)(ws + (size_t)N_ROWS * 4);        // 1024 floats
  unsigned* counts  = (unsigned*)(ws + (size_t)N_ROWS * 4 + 4096);
  float*    lossAcc = (float*)(ws + (size_t)N_ROWS * 4 + 8192);

  k0_init  <<<4,    256, 0, stream>>>(emb, eNormSq, counts, lossAcc);
  k1_argmin<<<512,  256, 0, stream>>>(x, emb, eNormSq, idxArr);
  k2_gather<<<1024, 256, 0, stream>>>(x, emb, idxArr, quant, counts, lossAcc);
  k3_zero  <<<2048, 256, 0, stream>>>(enc, (long)N_ROWS * NUM_EMB / 2);
  k4_scatter<<<N_ROWS / 256, 256, 0, stream>>>(idxArr, enc);
  k5_final <<<1,    256, 0, stream>>>(counts, lossAcc, outLoss, outPpl);
}